// MultiHeadAttention_to_attn_81990925681133
// MI455X (gfx1250) — compile-verified
//
#include <hip/hip_runtime.h>
#include <hip/hip_bf16.h>
#include <math.h>

typedef __attribute__((ext_vector_type(2))) float v2f;
typedef __attribute__((ext_vector_type(8))) float v8f;

#define HH 8
#define BB 256
#define NN 200
#define DD 128
#define KDIM 16
#define TRI_CNT 19900
#define QSTRIDE 132            // padded q row stride (floats) to avoid LDS bank conflicts
#define NROWS_PAD 208          // 13 * 16 (zero-padded rows 200..207)
#define NEGINF (-__builtin_inff())

// dynamic LDS partition (byte offsets)
#define SM_COMP  0             // 40000 floats: masked scores; also q staging (208*132 floats)
#define SM_Q     160000        // 208*16 floats
#define SM_K     173312        // 208*16 floats
#define SM_W     186624        // 2 * 128*16 floats (Wq | Wk)
#define SM_RED   203008        // 32 floats reduction scratch
#define SM_BYTES 203136

__global__ __launch_bounds__(256)
void MultiHeadAttention_to_attn_81990925681133_kernel(
    const float* __restrict__ qg,
    const unsigned char* __restrict__ che_g,   // jax bool -> 1 byte per element
    const int* __restrict__ exch,
    const float* __restrict__ wq,
    const float* __restrict__ wk,
    float* __restrict__ out)
{
  extern __shared__ char smem[];
  float* sComp = (float*)(smem + SM_COMP);
  float* sQin  = sComp;                         // q staging, stride QSTRIDE, 208 rows
  float* sQ    = (float*)(smem + SM_Q);
  float* sK    = (float*)(smem + SM_K);
  float* sW    = (float*)(smem + SM_W);
  float* sRed  = (float*)(smem + SM_RED);

  const int tid  = threadIdx.x;
  const int lane = tid & 31;
  const int wave = tid >> 5;
  const int l    = lane & 15;                   // lane within half-wave
  const int half = lane >> 4;                   // 0: lanes 0-15, 1: lanes 16-31
  const int wg   = blockIdx.x;                  // h*256 + b
  const int h    = wg >> 8;
  const int b    = wg & 255;

  // ---------- phase 1: stage q[b] (200x128, rows 200..207 zeroed) + Wq[h]/Wk[h] ----------
  {
    const float4* src = (const float4*)(qg + (size_t)b * NN * DD);
    for (int f = tid; f < (NN * DD) / 4; f += 256) {
      float4 v = src[f];
      int row = f >> 5;                         // 32 float4 per 128-wide row
      int c4  = (f & 31) << 2;
      float* d = sQin + row * QSTRIDE + c4;
      d[0] = v.x; d[1] = v.y; d[2] = v.z; d[3] = v.w;
    }
    // zero the 8 padding rows so projection A-loads need no bounds check
    for (int f = tid; f < (NROWS_PAD - NN) * QSTRIDE; f += 256)
      sQin[NN * QSTRIDE + f] = 0.0f;
    const float* wqh = wq + (size_t)h * DD * KDIM;
    const float* wkh = wk + (size_t)h * DD * KDIM;
    for (int f = tid; f < DD * KDIM; f += 256) {
      sW[f]        = wqh[f];
      sW[2048 + f] = wkh[f];
    }
  }
  __syncthreads();

  // ---------- phase 2: Q = q@Wq, K = q@Wk via v_wmma_f32_16x16x4_f32 ----------
  // 26 jobs = 13 row-tiles x {Q,K}; one 16x16 f32 accumulator per wave-job.
  for (int job = wave; job < 26; job += 8) {
    const int mat = job & 1;                    // 0 -> Q, 1 -> K
    const int r0  = (job >> 1) << 4;            // tile row base (0..192)
    const float* Wb = sW + mat * 2048;
    float* dst = mat ? sK : sQ;
    const int row = r0 + l;                     // always valid (staging zero-padded)
    v8f c = {};
    #pragma unroll 4
    for (int k = 0; k < DD; k += 4) {
      const int kb = k + 2 * half;              // A/B VGPR0 holds K=0 (lo lanes), K=2 (hi lanes)
      v2f a, bb;
      a.x  = sQin[row * QSTRIDE + kb];
      a.y  = sQin[row * QSTRIDE + kb + 1];
      bb.x = Wb[kb * KDIM + l];
      bb.y = Wb[(kb + 1) * KDIM + l];
      c = __builtin_amdgcn_wmma_f32_16x16x4_f32(false, a, false, bb, (short)0, c, false, false);
    }
    // D layout: VGPR j -> row j (lanes 0-15) / row j+8 (lanes 16-31), col = lane&15
    #pragma unroll
    for (int jj = 0; jj < 8; ++jj)
      dst[(r0 + jj + 8 * half) * KDIM + l] = c[jj];
  }
  __syncthreads();

  // ---------- phase 3: S = Q.K^T, tanh-rescale, masks -> sComp ----------
  const unsigned char* che = che_g + ((size_t)h * BB + b) * TRI_CNT;
  for (int job = wave; job < 169; job += 8) {
    const int ti = job / 13, tj = job - 13 * (job / 13);
    const int i0 = ti << 4, j0 = tj << 4;
    if ((j0 + 15 <= i0) || (i0 >= NN - 28)) {   // tile fully masked (lower-tri / last 28 rows)
      #pragma unroll
      for (int jj = 0; jj < 8; ++jj) {
        int i = i0 + jj + 8 * half, j = j0 + l;
        if (i < NN && j < NN) sComp[i * NN + j] = NEGINF;
      }
      continue;
    }
    v8f c = {};
    #pragma unroll
    for (int k = 0; k < KDIM; k += 4) {
      const int kb = k + 2 * half;
      v2f a, bb;
      a.x  = sQ[(i0 + l) * KDIM + kb];
      a.y  = sQ[(i0 + l) * KDIM + kb + 1];
      bb.x = sK[(j0 + l) * KDIM + kb];          // B = K^T : B[k][j] = K[j][k]
      bb.y = sK[(j0 + l) * KDIM + kb + 1];
      c = __builtin_amdgcn_wmma_f32_16x16x4_f32(false, a, false, bb, (short)0, c, false, false);
    }
    #pragma unroll
    for (int jj = 0; jj < 8; ++jj) {
      int i = i0 + jj + 8 * half, j = j0 + l;
      if (i >= NN || j >= NN) continue;
      float val;
      if (j <= i || i >= NN - 28 || j == 100) {
        val = NEGINF;
      } else {
        int tri = i * (NN - 1) - ((i * (i - 1)) >> 1) + (j - i - 1);
        val = che[tri] ? tanhf(0.25f * c[jj]) * 10.0f : NEGINF;
      }
      sComp[i * NN + j] = val;
    }
  }
  __syncthreads();

  // ---------- phase 3b: head-0 per-batch exchange-pair scatter ----------
  if (h == 0 && tid == 0) {
    int e0 = exch[2 * b], e1 = exch[2 * b + 1];
    sComp[e1 * NN + e0] = NEGINF;
    sComp[e0 * NN + e1] = NEGINF;
  }
  __syncthreads();

  // ---------- phase 4: flat softmax over 40000 elements ----------
  const float4* cv = (const float4*)sComp;      // 10000 float4
  float m = NEGINF;
  for (int f = tid; f < 10000; f += 256) {
    float4 v = cv[f];
    m = fmaxf(fmaxf(m, fmaxf(v.x, v.y)), fmaxf(v.z, v.w));
  }
  #pragma unroll
  for (int off = 16; off; off >>= 1) m = fmaxf(m, __shfl_xor(m, off, 32));
  if (lane == 0) sRed[wave] = m;
  __syncthreads();
  float gm = sRed[0];
  #pragma unroll
  for (int w = 1; w < 8; ++w) gm = fmaxf(gm, sRed[w]);
  __syncthreads();

  float s = 0.0f;
  for (int f = tid; f < 10000; f += 256) {
    float4 v = cv[f];
    s += expf(v.x - gm) + expf(v.y - gm) + expf(v.z - gm) + expf(v.w - gm);
  }
  #pragma unroll
  for (int off = 16; off; off >>= 1) s += __shfl_xor(s, off, 32);
  if (lane == 0) sRed[wave] = s;
  __syncthreads();
  float gs = 0.0f;
  #pragma unroll
  for (int w = 0; w < 8; ++w) gs += sRed[w];
  const float gl = gm + logf(gs);               // logsumexp

  // stream both outputs: log_softmax then softmax (concatenated)
  float4* outLog = (float4*)out + (size_t)wg * 10000;
  float4* outSm  = (float4*)out + (size_t)HH * BB * 10000 + (size_t)wg * 10000;
  for (int f = tid; f < 10000; f += 256) {
    float4 v = cv[f];
    float4 lo = make_float4(v.x - gl, v.y - gl, v.z - gl, v.w - gl);
    float4 sm = make_float4(expf(lo.x), expf(lo.y), expf(lo.z), expf(lo.w));
    outLog[f] = lo;
    outSm[f]  = sm;
  }
}

extern "C" void kernel_launch(void* const* d_in, const int* in_sizes, int n_in,
                              void* d_out, int out_size, void* d_ws, size_t ws_size,
                              hipStream_t stream) {
  (void)in_sizes; (void)n_in; (void)out_size; (void)d_ws; (void)ws_size;
  const float*         q    = (const float*)d_in[0];
  const unsigned char* che  = (const unsigned char*)d_in[1]; // bool array (1B elems)
  const int*           exch = (const int*)d_in[2];
  const float*         wq   = (const float*)d_in[3];
  const float*         wk   = (const float*)d_in[4];
  float*               out  = (float*)d_out;

  MultiHeadAttention_to_attn_81990925681133_kernel<<<dim3(HH * BB), dim3(256), SM_BYTES, stream>>>(
      q, che, exch, wq, wk, out);
}